// VectorQuantizer_15642270892432
// MI455X (gfx1250) — compile-verified
//
#include <hip/hip_runtime.h>
#include <hip/hip_bf16.h>

// Vector-quantizer for MI455X (gfx1250, wave32).
// Distances via V_WMMA_F32_16X16X4_F32: one wave computes a 16(token) x 512(codebook)
// score tile, 16 WMMAs per 16-wide column tile (K=64 in steps of 4), 32 tiles.
// 32 WMMA/token == exactly the 2*K*D FLOPs of the reference GEMM, no padding waste.
// B fragments are double-buffered in registers: tile t+1's 16 loads are issued while
// tile t's 16 WMMAs execute, so each wave hides its own L2 latency.

typedef float v2f __attribute__((ext_vector_type(2)));
typedef float v8f __attribute__((ext_vector_type(8)));

#define DIMS 64
#define NEMB 512
#define TILES (NEMB / 16)   // 32 column tiles of 16 embeddings

// ---------------------------------------------------------------- ||e_k||^2
__global__ void vq_enorm_kernel(const float* __restrict__ E, float* __restrict__ enorm) {
    int k = blockIdx.x * blockDim.x + threadIdx.x;
    if (k < NEMB) {
        const float* row = E + (size_t)k * DIMS;
        float s = 0.f;
        #pragma unroll
        for (int d = 0; d < DIMS; ++d) { float v = row[d]; s += v * v; }
        enorm[k] = s;
    }
}

// ---------------------------------------------------------------- main kernel
// block = 256 threads = 8 waves; each wave owns 16 tokens -> 128 tokens/block.
__global__ __launch_bounds__(256)
void vq_main_kernel(const float* __restrict__ X, const float* __restrict__ E,
                    const float* __restrict__ enorm,
                    float* __restrict__ qOut, float* __restrict__ idxOut,
                    float* __restrict__ blockPartial, int nTokens) {
    const int lane = threadIdx.x & 31;
    const int lr   = lane & 15;   // row/col within half (A: M, B: N)
    const int h    = lane >> 4;   // half-wave
    const int waveGlobal = blockIdx.x * (blockDim.x >> 5) + (threadIdx.x >> 5);
    const int tokenBase  = waveGlobal * 16;
    const bool active = (tokenBase + 16) <= nTokens;   // wave-uniform -> EXEC stays all-1s

    float lossAcc = 0.f;

    if (active) {
        // A fragments: 16x4 fp32 layout -> reg j, half h holds K = j + 2h, M = lr.
        // Step s covers K = 4s..4s+3, so this lane needs X[lr][4s+2h], X[lr][4s+2h+1].
        v2f a[16];
        const float* xrow = X + (size_t)(tokenBase + lr) * DIMS;
        #pragma unroll
        for (int s = 0; s < 16; ++s)
            a[s] = *(const v2f*)(xrow + 4 * s + 2 * h);

        float bestVal[8];
        int   bestIdx[8];
        #pragma unroll
        for (int j = 0; j < 8; ++j) { bestVal[j] = __builtin_inff(); bestIdx[j] = 0; }

        // B fragment loader: embedding rows nb..nb+15, this lane's K slices.
        auto loadB = [&](v2f (&buf)[16], int nb) {
            const float* erow = E + (size_t)(nb + lr) * DIMS;   // B: N = lr, K = j + 2h
            #pragma unroll
            for (int s = 0; s < 16; ++s)
                buf[s] = *(const v2f*)(erow + 4 * s + 2 * h);
        };

        // 16 WMMAs over K=64 in two independent accumulation chains, then argmin update.
        auto processTile = [&](const v2f (&buf)[16], int nb) {
            v8f acc0 = {};
            v8f acc1 = {};
            #pragma unroll
            for (int s = 0; s < 8; ++s) {
                acc0 = __builtin_amdgcn_wmma_f32_16x16x4_f32(
                           false, a[2 * s],     false, buf[2 * s],     (short)0, acc0, false, false);
                acc1 = __builtin_amdgcn_wmma_f32_16x16x4_f32(
                           false, a[2 * s + 1], false, buf[2 * s + 1], (short)0, acc1, false, false);
            }
            // C/D layout: lane's 8 slots all have N = nb + lr (M = j + 8h).
            const float en    = enorm[nb + lr];
            const int   myIdx = nb + lr;
            #pragma unroll
            for (int j = 0; j < 8; ++j) {
                float d = en - 2.0f * (acc0[j] + acc1[j]);  // ||x||^2 const per row: dropped
                if (d < bestVal[j]) { bestVal[j] = d; bestIdx[j] = myIdx; }
            }
        };

        // Software pipeline: while tile t's WMMAs run, tile t+1's B loads are in flight.
        v2f bA[16], bB[16];
        loadB(bA, 0);
        for (int tt = 0; tt < TILES; tt += 2) {
            loadB(bB, (tt + 1) * 16);
            processTile(bA, tt * 16);
            if (tt + 2 < TILES) loadB(bA, (tt + 2) * 16);
            processTile(bB, (tt + 1) * 16);
        }

        // argmin across the 16 lanes of each half (masks < 16 stay within the half).
        // Tie-break to smaller index == jnp.argmin first-occurrence semantics.
        #pragma unroll
        for (int m = 1; m < 16; m <<= 1) {
            #pragma unroll
            for (int j = 0; j < 8; ++j) {
                float ov = __shfl_xor(bestVal[j], m, 32);
                int   oi = __shfl_xor(bestIdx[j], m, 32);
                if (ov < bestVal[j] || (ov == bestVal[j] && oi < bestIdx[j])) {
                    bestVal[j] = ov; bestIdx[j] = oi;
                }
            }
        }

        // indices (as float, reference dtype is promoted on concat)
        if (lr == 0) {
            #pragma unroll
            for (int j = 0; j < 8; ++j)
                idxOut[tokenBase + j + 8 * h] = (float)bestIdx[j];
        }

        // gather rows, write quantized (straight-through output == q numerically),
        // accumulate (q-x)^2. Each half handles its 8 rows; 16 lanes x float4 = full row x4.
        #pragma unroll
        for (int j = 0; j < 8; ++j) {
            const int row = tokenBase + j + 8 * h;
            const int idx = bestIdx[j];                      // uniform across the half
            const float4 q = *(const float4*)(E + (size_t)idx * DIMS + 4 * lr);
            const float4 x = *(const float4*)(X + (size_t)row * DIMS + 4 * lr);
            *(float4*)(qOut + (size_t)row * DIMS + 4 * lr) = q;
            float dx = q.x - x.x, dy = q.y - x.y, dz = q.z - x.z, dw = q.w - x.w;
            lossAcc += dx * dx + dy * dy + dz * dz + dw * dw;
        }
    }

    // deterministic block reduction of the loss partials
    __shared__ float red[256];
    red[threadIdx.x] = lossAcc;
    __syncthreads();
    #pragma unroll
    for (int s = 128; s > 0; s >>= 1) {
        if (threadIdx.x < (unsigned)s) red[threadIdx.x] += red[threadIdx.x + s];
        __syncthreads();
    }
    if (threadIdx.x == 0) blockPartial[blockIdx.x] = red[0];
}

// ---------------------------------------------------------------- finalize
__global__ void vq_finalize_kernel(const float* __restrict__ partials,
                                   float* __restrict__ lossOut,
                                   int nPartials, float scale) {
    __shared__ float red[256];
    float s = 0.f;
    for (int i = threadIdx.x; i < nPartials; i += 256) s += partials[i];  // fixed order
    red[threadIdx.x] = s;
    __syncthreads();
    #pragma unroll
    for (int st = 128; st > 0; st >>= 1) {
        if (threadIdx.x < (unsigned)st) red[threadIdx.x] += red[threadIdx.x + st];
        __syncthreads();
    }
    if (threadIdx.x == 0) *lossOut = red[0] * scale;
}

extern "C" void kernel_launch(void* const* d_in, const int* in_sizes, int n_in,
                              void* d_out, int out_size, void* d_ws, size_t ws_size,
                              hipStream_t stream) {
    const float* X = (const float*)d_in[0];   // [N, 64]
    const float* E = (const float*)d_in[1];   // [512, 64]
    const int nTok = in_sizes[0] / DIMS;

    float* qOut    = (float*)d_out;                          // [N*64]
    float* lossOut = (float*)d_out + (size_t)nTok * DIMS;    // [1]
    float* idxOut  = lossOut + 1;                            // [N]

    float* enorm    = (float*)d_ws;        // 512 floats
    float* partials = enorm + NEMB;        // one per block

    const int grid = (nTok + 127) / 128;   // 128 tokens per 256-thread block

    vq_enorm_kernel<<<(NEMB + 255) / 256, 256, 0, stream>>>(E, enorm);
    vq_main_kernel<<<grid, 256, 0, stream>>>(X, E, enorm, qOut, idxOut, partials, nTok);
    vq_finalize_kernel<<<1, 256, 0, stream>>>(partials, lossOut, grid,
                                              1.25f / ((float)nTok * (float)DIMS));
}